// GraphAttentionLayer_66649302499990
// MI455X (gfx1250) — compile-verified
//
#include <hip/hip_runtime.h>
#include <hip/hip_bf16.h>
#include <math.h>

// GAT fused kernel set for MI455X (gfx1250, wave32, WMMA).
// Shapes: B=8, N=2048, Fin=768, Fout=32.

typedef __attribute__((ext_vector_type(2))) float v2f;
typedef __attribute__((ext_vector_type(8))) float v8f;

#define BATCH 8
#define NN    2048
#define FIN   768
#define FOUT  32
#define ALPHA 0.3f
#define EPS_  1e-5f
#define ROWS  (BATCH * NN)        // 16384 flattened (b,n) rows

// workspace layout (floats): h[ROWS*FOUT] | s1[ROWS] | s2[ROWS]
#define H_SZ (ROWS * FOUT)

// ---------------------------------------------------------------------------
// K1: h = feat @ W   (M=16384, K=768, N=32) with V_WMMA_F32_16X16X4_F32.
// One wave owns a 16-row tile and both 16-wide Fout halves (2 accumulators).
// A layout (16x4 f32): lane L: row M=(L&15); VGPR0 holds K=(L>>4)*2, VGPR1 K+1.
// ---------------------------------------------------------------------------
__global__ __launch_bounds__(256) void gat_h_kernel(const float* __restrict__ feat,
                                                    const float* __restrict__ W,
                                                    float* __restrict__ h) {
  const int lane = threadIdx.x & 31;
  const int wave = threadIdx.x >> 5;
  const int tile = blockIdx.x * 8 + wave;      // 0..1023
  const int m0 = tile * 16;
  const int r  = lane & 15;
  const int hi = lane >> 4;
  const int kb = hi * 2;

  v8f c0 = {};
  v8f c1 = {};
  const float* arow = feat + (size_t)(m0 + r) * FIN + kb;
  for (int k = 0; k < FIN; k += 4) {
    v2f a = *(const v2f*)(arow + k);                 // feat[m0+r][k+kb .. k+kb+1]
    const float* wp = W + (size_t)(k + kb) * FOUT + r;
    v2f b0, b1;
    b0.x = wp[0];        b0.y = wp[FOUT];            // W[k+kb][r],    W[k+kb+1][r]
    b1.x = wp[16];       b1.y = wp[FOUT + 16];       // W[k+kb][r+16], W[k+kb+1][r+16]
    c0 = __builtin_amdgcn_wmma_f32_16x16x4_f32(false, a, false, b0, (short)0, c0, false, false);
    c1 = __builtin_amdgcn_wmma_f32_16x16x4_f32(false, a, false, b1, (short)0, c1, false, false);
  }
  // C/D layout: lane = N(=f), VGPR p -> M = p + hi*8
  float* hrow = h + (size_t)(m0 + hi * 8) * FOUT;
  #pragma unroll
  for (int p = 0; p < 8; ++p) {
    hrow[p * FOUT + r]      = c0[p];
    hrow[p * FOUT + 16 + r] = c1[p];
  }
}

// ---------------------------------------------------------------------------
// K2: s1[m] = h[m,:]·a[:32], s2[m] = h[m,:]·a[32:64]; wave-per-row, wave32
// butterfly reduction.
// ---------------------------------------------------------------------------
__global__ __launch_bounds__(256) void gat_s_kernel(const float* __restrict__ h,
                                                    const float* __restrict__ a,
                                                    float* __restrict__ s1,
                                                    float* __restrict__ s2) {
  const int lane = threadIdx.x & 31;
  const int wave = threadIdx.x >> 5;
  const int m = blockIdx.x * 8 + wave;
  const float hv = h[(size_t)m * FOUT + lane];
  float v1 = hv * a[lane];
  float v2 = hv * a[FOUT + lane];
  #pragma unroll
  for (int off = 16; off > 0; off >>= 1) {
    v1 += __shfl_xor(v1, off, 32);
    v2 += __shfl_xor(v2, off, 32);
  }
  if (lane == 0) { s1[m] = v1; s2[m] = v2; }
}

// ---------------------------------------------------------------------------
// K3: fused masked-softmax-attention + (attn @ h).
// Block = 128 threads = 4 waves; block owns (b, 16-row tile); each wave owns a
// 512-wide j range. Un-normalized single-pass softmax (no max subtraction; e is
// bounded << 88 so exp(e) is f32-safe; masked entries are exactly 0, matching
// exp(NEG - max) underflow in the reference). Partial accumulators combine by
// plain summation across waves. Pre-norm result written to d_out.
// ---------------------------------------------------------------------------
__global__ __launch_bounds__(128) void gat_attn_kernel(const int* __restrict__ adj,
                                                       const float* __restrict__ h,
                                                       const float* __restrict__ s1,
                                                       const float* __restrict__ s2,
                                                       float* __restrict__ out_pre) {
  __shared__ __align__(16) float lds_s2[4 * 512];
  __shared__ float red[4][16][FOUT];
  __shared__ float zred[4][32];

  const int lane = threadIdx.x & 31;
  const int wave = threadIdx.x >> 5;
  const int b  = blockIdx.y;
  const int i0 = blockIdx.x * 16;
  const int r  = lane & 15;
  const int hi = lane >> 4;
  const int kb = hi * 2;
  const int jw0 = wave * 512;

  // stage this wave's s2 chunk into LDS (wave-private region, no barrier needed)
  {
    const float4* src = (const float4*)(s2 + (size_t)b * NN + jw0);
    float4* dst = (float4*)(lds_s2 + wave * 512);
    #pragma unroll
    for (int q = 0; q < 4; ++q) dst[lane + q * 32] = src[lane + q * 32];
  }

  const float s1v = s1[(size_t)b * NN + i0 + r];
  const int*   adjrow = adj + ((size_t)b * NN + (i0 + r)) * NN + kb;
  const float* hbase  = h + ((size_t)b * NN) * FOUT;
  const float* s2l    = lds_s2 + wave * 512 + kb;

  v8f c0 = {};
  v8f c1 = {};
  float zacc = 0.0f;

  for (int jj = 0; jj < 512; jj += 4) {
    const int j = jw0 + jj;
    if ((jj & 63) == 0)
      __builtin_prefetch(adjrow + j + 256, 0, 0);    // stream adj ahead (global_prefetch_b8)

    const int2 am = *(const int2*)(adjrow + j);      // adj[b][i0+r][j+kb], [j+kb+1]
    float e0 = s1v + s2l[jj];
    float e1 = s1v + s2l[jj + 1];
    e0 = (e0 >= 0.0f) ? e0 : ALPHA * e0;
    e1 = (e1 >= 0.0f) ? e1 : ALPHA * e1;
    const float p0 = (am.x > 0) ? __expf(e0) : 0.0f;
    const float p1 = (am.y > 0) ? __expf(e1) : 0.0f;
    zacc += p0 + p1;

    v2f A; A.x = p0; A.y = p1;                       // A tile: attn[16 x 4]
    const float* hb = hbase + (size_t)(j + kb) * FOUT;
    v2f B0, B1;                                      // B tiles: h[4 x 16] per half
    B0.x = hb[r];        B0.y = hb[FOUT + r];
    B1.x = hb[16 + r];   B1.y = hb[FOUT + 16 + r];
    c0 = __builtin_amdgcn_wmma_f32_16x16x4_f32(false, A, false, B0, (short)0, c0, false, false);
    c1 = __builtin_amdgcn_wmma_f32_16x16x4_f32(false, A, false, B1, (short)0, c1, false, false);
  }

  // dump partials to LDS and combine across the 4 waves
  #pragma unroll
  for (int p = 0; p < 8; ++p) {
    red[wave][p + hi * 8][r]      = c0[p];
    red[wave][p + hi * 8][16 + r] = c1[p];
  }
  zred[wave][lane] = zacc;
  __syncthreads();

  // 128 threads -> 512 outputs (16 rows x 32 feats), 4 consecutive feats each
  const int base = threadIdx.x * 4;
  const int il = base >> 5;
  const int f0 = base & 31;
  float Z = 0.0f;
  #pragma unroll
  for (int w = 0; w < 4; ++w) Z += zred[w][il] + zred[w][il + 16];
  const float inv = 1.0f / Z;
  float* op = out_pre + ((size_t)b * NN + i0 + il) * FOUT + f0;
  #pragma unroll
  for (int q = 0; q < 4; ++q) {
    const float v = red[0][il][f0 + q] + red[1][il][f0 + q] +
                    red[2][il][f0 + q] + red[3][il][f0 + q];
    op[q] = v * inv;
  }
}

// ---------------------------------------------------------------------------
// K4: per-node LayerNorm over (B, Fout) = 256 values + ELU, in place on d_out.
// Wave-per-node: lane = f, loop q = b; butterfly reduce sum & sumsq.
// ---------------------------------------------------------------------------
__global__ __launch_bounds__(256) void gat_norm_kernel(float* __restrict__ out,
                                                       const float* __restrict__ gamma,
                                                       const float* __restrict__ beta) {
  const int lane = threadIdx.x & 31;
  const int wave = threadIdx.x >> 5;
  const int n = blockIdx.x * 8 + wave;
  float vals[BATCH];
  float sum = 0.0f, sq = 0.0f;
  #pragma unroll
  for (int q = 0; q < BATCH; ++q) {
    vals[q] = out[((size_t)q * NN + n) * FOUT + lane];
    sum += vals[q];
    sq  += vals[q] * vals[q];
  }
  #pragma unroll
  for (int off = 16; off > 0; off >>= 1) {
    sum += __shfl_xor(sum, off, 32);
    sq  += __shfl_xor(sq,  off, 32);
  }
  const float mean = sum * (1.0f / 256.0f);
  const float var  = sq  * (1.0f / 256.0f) - mean * mean;
  const float scl  = rsqrtf(var + EPS_) * gamma[n];
  const float bb   = beta[n];
  #pragma unroll
  for (int q = 0; q < BATCH; ++q) {
    const float y = (vals[q] - mean) * scl + bb;
    out[((size_t)q * NN + n) * FOUT + lane] = (y > 0.0f) ? y : expm1f(y);
  }
}

// ---------------------------------------------------------------------------
extern "C" void kernel_launch(void* const* d_in, const int* in_sizes, int n_in,
                              void* d_out, int out_size, void* d_ws, size_t ws_size,
                              hipStream_t stream) {
  const float* feat  = (const float*)d_in[0];   // (8,2048,768) f32
  const int*   adj   = (const int*)  d_in[1];   // (8,2048,2048) i32
  const float* W     = (const float*)d_in[2];   // (768,32) f32
  const float* a     = (const float*)d_in[3];   // (64,1) f32
  const float* gamma = (const float*)d_in[4];   // (2048,) f32
  const float* beta  = (const float*)d_in[5];   // (2048,) f32

  float* ws = (float*)d_ws;
  float* h  = ws;                 // ROWS*FOUT
  float* s1 = ws + H_SZ;          // ROWS
  float* s2 = ws + H_SZ + ROWS;   // ROWS
  float* out = (float*)d_out;     // (8,2048,32) f32; used as pre-norm staging too

  gat_h_kernel   <<<128, 256, 0, stream>>>(feat, W, h);
  gat_s_kernel   <<<2048, 256, 0, stream>>>(h, a, s1, s2);
  gat_attn_kernel<<<dim3(128, BATCH), 128, 0, stream>>>(adj, h, s1, s2, out);
  gat_norm_kernel<<<256, 256, 0, stream>>>(out, gamma, beta);
}